// MultiBoxLoss_35021163331610
// MI455X (gfx1250) — compile-verified
//
#include <hip/hip_runtime.h>
#include <math.h>

#define NPRIORS  125
#define NCLASSES 21
#define THRESH   0.3f

typedef __attribute__((ext_vector_type(2))) float v2f;
typedef __attribute__((ext_vector_type(8))) float v8f;

__device__ __forceinline__ float smooth_l1(float d) {
  float ad = fabsf(d);
  return (ad < 1.0f) ? (0.5f * d * d) : (ad - 0.5f);
}

// Tree reduction over 128 threads; fixed order -> deterministic.
__device__ __forceinline__ float block_reduce_sum_128(float v, float* s, int p) {
  s[p] = v;
  __syncthreads();
#pragma unroll
  for (int st = 64; st > 0; st >>= 1) {
    if (p < st) s[p] += s[p + st];
    __syncthreads();
  }
  float r = s[0];
  __syncthreads();
  return r;
}

// One block (128 threads) per batch item; thread p handles prior p (125 real, 3 pad).
__global__ __launch_bounds__(128) void mbx_stage1(
    const float* __restrict__ locs, const float* __restrict__ scores,
    const float* __restrict__ target, const float* __restrict__ priors,
    float* __restrict__ ws, int batch)
{
  const int b = blockIdx.x;
  const int p = threadIdx.x;

  __shared__ float sbox[8];
  __shared__ float s_val[128];
  __shared__ int   s_arg[128];
  __shared__ float s_conf[128];

  if (p < 8) sbox[p] = target[(size_t)b * 8 + p];
  __syncthreads();

  const float bx = sbox[0], by = sbox[1], bw = sbox[2], bh = sbox[3];
  const float tsin = sbox[5], tcos = sbox[6];
  const float flabel = sbox[7];

  const float* lp = locs   + ((size_t)b * NPRIORS + (p < NPRIORS ? p : 0)) * 6;
  const float* sp = scores + ((size_t)b * NPRIORS + (p < NPRIORS ? p : 0)) * NCLASSES;
  if (p < NPRIORS) {
    __builtin_prefetch(sp, 0, 0);   // global_prefetch_b8
    __builtin_prefetch(lp, 0, 0);
  }

  // ---- IoU vs prior p ----
  float iou = -1.0f;
  float px = 0.f, py = 0.f, pw = 1.f, ph = 1.f;
  if (p < NPRIORS) {
    float4 pr = reinterpret_cast<const float4*>(priors)[p];
    px = pr.x; py = pr.y; pw = pr.z; ph = pr.w;
    float iw = fminf(bx + 0.5f * bw, px + 0.5f * pw) - fmaxf(bx - 0.5f * bw, px - 0.5f * pw);
    float ih = fminf(by + 0.5f * bh, py + 0.5f * ph) - fmaxf(by - 0.5f * bh, py - 0.5f * ph);
    iw = fmaxf(iw, 0.0f); ih = fmaxf(ih, 0.0f);
    float inter = iw * ih;
    iou = inter / (bw * bh + pw * ph - inter);
  }

  // ---- argmax (first-max tie-break, like jnp.argmax) ----
  s_val[p] = iou; s_arg[p] = p;
  __syncthreads();
#pragma unroll
  for (int st = 64; st > 0; st >>= 1) {
    if (p < st) {
      float vr = s_val[p + st]; int ar = s_arg[p + st];
      float vl = s_val[p];      int al = s_arg[p];
      if (vr > vl || (vr == vl && ar < al)) { s_val[p] = vr; s_arg[p] = ar; }
    }
    __syncthreads();
  }
  const int best = s_arg[0];
  __syncthreads();

  const float ov  = (p == best) ? 1.0f : iou;
  const bool  pos = (p < NPRIORS) && !(ov < THRESH);
  const int   cls = pos ? (int)flabel : 0;

  // ---- per-prior losses ----
  float loc_sum = 0.0f, ang_sum = 0.0f, conf_all = 0.0f;
  if (p < NPRIORS) {
    float l0 = lp[0], l1 = lp[1], l2 = lp[2], l3 = lp[3], l4 = lp[4], l5 = lp[5];
    if (pos) {
      float g0 = (bx - px) * 10.0f / pw;
      float g1 = (by - py) * 10.0f / ph;
      float g2 = logf(bw / pw) * 5.0f;
      float g3 = logf(bh / ph) * 5.0f;
      loc_sum = smooth_l1(l0 - g0) + smooth_l1(l1 - g1) +
                smooth_l1(l2 - g2) + smooth_l1(l3 - g3);
      float da0 = l4 - tsin, da1 = l5 - tcos;
      ang_sum = da0 * da0 + da1 * da1;
    }
    // logsumexp over 21 classes (register resident, single global pass)
    float v[NCLASSES];
#pragma unroll
    for (int c = 0; c < NCLASSES; ++c) v[c] = sp[c];
    float m = v[0];
#pragma unroll
    for (int c = 1; c < NCLASSES; ++c) m = fmaxf(m, v[c]);
    float se = 0.0f;
#pragma unroll
    for (int c = 0; c < NCLASSES; ++c) se += expf(v[c] - m);
    conf_all = (m + logf(se)) - v[cls];   // >= 0 always
  }

  const float npos_f   = block_reduce_sum_128(pos ? 1.0f : 0.0f, s_val, p);
  const float loc_tot  = block_reduce_sum_128(loc_sum, s_val, p);
  const float ang_tot  = block_reduce_sum_128(ang_sum, s_val, p);
  const float cpos_tot = block_reduce_sum_128(pos ? conf_all : 0.0f, s_val, p);

  // ---- hard-negative mining: bitonic ascending sort of conf_neg (padded with 0) ----
  s_conf[p] = (p < NPRIORS && !pos) ? conf_all : 0.0f;
  __syncthreads();
  for (int k = 2; k <= 128; k <<= 1) {
    for (int j = k >> 1; j > 0; j >>= 1) {
      int ixj = p ^ j;
      if (ixj > p) {
        float a = s_conf[p], c2 = s_conf[ixj];
        bool asc = ((p & k) == 0);
        if ((a > c2) == asc) { s_conf[p] = c2; s_conf[ixj] = a; }
      }
      __syncthreads();
    }
  }
  int npos = (int)npos_f;
  int K = 3 * npos; if (K > 128) K = 128;   // padded zeros contribute 0 -> identical sum
  float contrib = (p >= 128 - K) ? s_conf[p] : 0.0f;
  const float hard_tot = block_reduce_sum_128(contrib, s_val, p);

  if (p == 0) {
    ws[0 * (size_t)batch + b] = npos_f;
    ws[1 * (size_t)batch + b] = loc_tot;
    ws[2 * (size_t)batch + b] = ang_tot;
    ws[3 * (size_t)batch + b] = cpos_tot;
    ws[4 * (size_t)batch + b] = hard_tot;
  }
}

// Stage 2: one wave per quantity; deterministic f32 WMMA reduction.
// With B == all-ones, D[m,n] = sum_k A[m,k]; sum(D) == 16 * sum(A) exactly scalable by 2^-4.
__global__ __launch_bounds__(256) void mbx_stage2(const float* __restrict__ ws,
                                                  float* __restrict__ out, int nitems)
{
  __shared__ float s_tot[5];
  const int wave = threadIdx.x >> 5;   // wave-uniform branch keeps EXEC all-ones for WMMA
  const int lane = threadIdx.x & 31;

  if (wave < 5) {
    const float* arr = ws + (size_t)wave * nitems;
    v8f c0 = {}, c1 = {}, c2 = {}, c3 = {};
    v2f ones; ones.x = 1.0f; ones.y = 1.0f;
    const int nchunk = nitems >> 6;     // 64 f32 per WMMA A-operand
    int i = 0;
    for (; i + 4 <= nchunk; i += 4) {   // 4 independent accumulators break the RAW chain
      const float* p0 = arr + (size_t)(i + 0) * 64;
      const float* p1 = arr + (size_t)(i + 1) * 64;
      const float* p2 = arr + (size_t)(i + 2) * 64;
      const float* p3 = arr + (size_t)(i + 3) * 64;
      v2f a0, a1, a2, a3;
      a0.x = p0[lane]; a0.y = p0[lane + 32];
      a1.x = p1[lane]; a1.y = p1[lane + 32];
      a2.x = p2[lane]; a2.y = p2[lane + 32];
      a3.x = p3[lane]; a3.y = p3[lane + 32];
      c0 = __builtin_amdgcn_wmma_f32_16x16x4_f32(false, a0, false, ones, (short)0, c0, false, false);
      c1 = __builtin_amdgcn_wmma_f32_16x16x4_f32(false, a1, false, ones, (short)0, c1, false, false);
      c2 = __builtin_amdgcn_wmma_f32_16x16x4_f32(false, a2, false, ones, (short)0, c2, false, false);
      c3 = __builtin_amdgcn_wmma_f32_16x16x4_f32(false, a3, false, ones, (short)0, c3, false, false);
    }
    for (; i < nchunk; ++i) {
      const float* p0 = arr + (size_t)i * 64;
      v2f a0; a0.x = p0[lane]; a0.y = p0[lane + 32];
      c0 = __builtin_amdgcn_wmma_f32_16x16x4_f32(false, a0, false, ones, (short)0, c0, false, false);
    }
    v8f c;
#pragma unroll
    for (int j = 0; j < 8; ++j) c[j] = c0[j] + c1[j] + c2[j] + c3[j];
    float s = 0.0f;
#pragma unroll
    for (int j = 0; j < 8; ++j) s += c[j];
#pragma unroll
    for (int off = 16; off > 0; off >>= 1) s += __shfl_down(s, off, 32);
    if (lane == 0) {
      float total = s * (1.0f / 16.0f);     // exact: divide replicated row sums
      int rem = nitems & 63;                 // scalar tail (0 for 16384)
      const float* tail = arr + (size_t)(nitems - rem);
      for (int t = 0; t < rem; ++t) total += tail[t];
      s_tot[wave] = total;
    }
  }
  __syncthreads();
  if (threadIdx.x == 0) {
    float npos       = s_tot[0];
    float conf_loss  = (s_tot[4] + s_tot[3]) / npos;
    float loc_loss   = s_tot[1] / (npos * 4.0f);
    float angle_loss = 25.0f * s_tot[2] / (npos * 2.0f);
    out[0] = conf_loss;
    out[1] = loc_loss;
    out[2] = angle_loss;
    out[3] = conf_loss + loc_loss + angle_loss;   // ALPHA = 1
  }
}

extern "C" void kernel_launch(void* const* d_in, const int* in_sizes, int n_in,
                              void* d_out, int out_size, void* d_ws, size_t ws_size,
                              hipStream_t stream)
{
  const float* locs   = (const float*)d_in[0];  // (B, 125, 6)
  const float* scores = (const float*)d_in[1];  // (B, 125, 21)
  const float* target = (const float*)d_in[2];  // (B, 1, 8)
  const float* priors = (const float*)d_in[3];  // (125, 4)
  float* ws  = (float*)d_ws;                    // 5 * B floats of partials
  float* out = (float*)d_out;                   // 4 floats

  const int batch = in_sizes[2] / 8;            // 16384
  mbx_stage1<<<batch, 128, 0, stream>>>(locs, scores, target, priors, ws, batch);
  mbx_stage2<<<1, 256, 0, stream>>>(ws, out, batch);
}